// GCN_69733089018382
// MI455X (gfx1250) — compile-verified
//
#include <hip/hip_runtime.h>
#include <hip/hip_bf16.h>

// ---------------------------------------------------------------------------
// GCN on gfx1250 (MI455X).
//  - Node-feature GEMMs (50000x128 @ 128x128, fp32) run on the WMMA pipe with
//    V_WMMA_F32_16X16X4_F32 (fp32 in / fp32 accum, matches reference precision).
//  - W is staged in LDS as packed float2 K-pairs with a parity rotation so
//    every B fragment is one bank-conflict-free ds_load_b64.
//  - A fragments: row index clamped once (OOB rows never stored), one
//    unconditional global_load_b64 per K-step -> no exec-mask churn.
//  - Edge aggregation: coalesced gather + global_atomic_add_f32 scatter,
//    1 feature per thread (128-thread blocks), L2-resident (25.6MB << 192MB L2).
//  - Segment-max pooling uses u32 atomicMax on float bits (inputs >= 0 after
//    ReLU, so the unsigned compare equals the float compare).
// ---------------------------------------------------------------------------

typedef __attribute__((ext_vector_type(2))) float v2f;
typedef __attribute__((ext_vector_type(8))) float v8f;

#define FEAT 128

// ---------------- utility kernels ----------------
__global__ void k_fill(float* __restrict__ p, float v, int n) {
    int i = blockIdx.x * blockDim.x + threadIdx.x;
    if (i < n) p[i] = v;
}

__global__ void k_deg_edges(const int* __restrict__ dst, float* __restrict__ deg, int e) {
    int i = blockIdx.x * blockDim.x + threadIdx.x;
    if (i < e) atomicAdd(&deg[dst[i]], 1.0f);
}

__global__ void k_dinv(const float* __restrict__ deg, float* __restrict__ dinv, int n) {
    int i = blockIdx.x * blockDim.x + threadIdx.x;
    if (i < n) {
        float d = deg[i];
        dinv[i] = d > 0.0f ? rsqrtf(d) : 0.0f;
    }
}

__global__ void k_norm(const int* __restrict__ src, const int* __restrict__ dst,
                       const float* __restrict__ dinv, float* __restrict__ norm, int e) {
    int i = blockIdx.x * blockDim.x + threadIdx.x;
    if (i < e) norm[i] = dinv[src[i]] * dinv[dst[i]];
}

// out[node][f] = bias[f] + hw[node][f] * dinv[node]^2   (bias + self-loop term)
__global__ void k_bias_selfloop(const float* __restrict__ hw, const float* __restrict__ dinv,
                                const float* __restrict__ bias, float* __restrict__ out,
                                int total) {
    int i = blockIdx.x * blockDim.x + threadIdx.x;
    if (i < total) {
        int f = i & (FEAT - 1);
        int node = i >> 7;
        float di = dinv[node];
        out[i] = bias[f] + hw[i] * di * di;
    }
}

// one 128-thread block handles edges with grid stride; thread = feature.
// src/dst/norm loads are block-uniform -> scalarized to SMEM by the compiler.
__global__ void __launch_bounds__(FEAT)
k_scatter(const float* __restrict__ hw, const int* __restrict__ src,
          const int* __restrict__ dst, const float* __restrict__ norm,
          float* __restrict__ out, int e) {
    int f = threadIdx.x;
    for (int ed = blockIdx.x; ed < e; ed += gridDim.x) {
        int s = src[ed];
        int d = dst[ed];
        float v = hw[(size_t)s * FEAT + f] * norm[ed];
        atomicAdd(&out[(size_t)d * FEAT + f], v);
    }
}

__global__ void k_relu(float* __restrict__ p, int n) {
    int i = blockIdx.x * blockDim.x + threadIdx.x;
    if (i < n) p[i] = fmaxf(p[i], 0.0f);
}

// ---------------- fp32 WMMA GEMM:  OUT[N x 128] = A[N x 128] @ W[128 x 128] --------------
// Block = 256 threads (8 wave32).  Each block computes 128 output rows; each
// wave computes a 16x128 strip with 8 v8f accumulators, iterating K in steps
// of 4 via V_WMMA_F32_16X16X4_F32.
//
// LDS layout for W: element (kp, c) = {W[2kp][c], W[2kp+1][c]} stored at
// sW2[kp*128 + ((c + 16*(kp&1)) & 127)].  A WMMA B-fragment read uses
// kp = k0/2 + half (so kp parity == lane half), giving one ds_load_b64 per
// fragment with the two lane halves on disjoint 32-bank sets (conflict-free).
__global__ void __launch_bounds__(256)
k_gemm128(const float* __restrict__ A, const float* __restrict__ W,
          float* __restrict__ OUT, int nrows) {
    __shared__ v2f sW2[64 * FEAT];  // 64 KB

    const int tid = threadIdx.x;
    // stage W: 8192 float2 elements; both global reads coalesced per 128-thread group
    for (int idx = tid; idx < 64 * FEAT; idx += 256) {
        const int kp = idx >> 7;
        const int c  = idx & 127;
        v2f w;
        w.x = W[(2 * kp) * FEAT + c];
        w.y = W[(2 * kp + 1) * FEAT + c];
        sW2[kp * FEAT + ((c + ((kp & 1) << 4)) & 127)] = w;
    }
    __syncthreads();

    const int wave = tid >> 5;
    const int lane = tid & 31;
    const int half = lane >> 4;   // K-pair selector (== kp parity)
    const int l16  = lane & 15;   // M for A frag, N for B/C/D frags

    // A fragment source row; clamp instead of predicate: OOB rows only feed
    // output rows that are never stored.
    const int rowA = blockIdx.x * 128 + wave * 16 + l16;
    const int rowC = rowA < nrows ? rowA : (nrows - 1);
    const float* Arow = A + (size_t)rowC * FEAT + 2 * half;

    const int cbase = l16 + (half << 4);  // column rotation base for B reads

    v8f acc[8] = {};  // 8 column tiles of 16

    for (int k0 = 0; k0 < FEAT; k0 += 4) {
        // A 16x4 frag: lanes 0-15 carry K={k0,k0+1}, lanes 16-31 K={k0+2,k0+3}
        const v2f a = *(const v2f*)(Arow + k0);
        const v2f* srow = sW2 + ((k0 >> 1) + half) * FEAT;
#pragma unroll
        for (int n = 0; n < 8; ++n) {
            const v2f b = srow[(n * 16 + cbase) & 127];
            acc[n] = __builtin_amdgcn_wmma_f32_16x16x4_f32(
                false, a, false, b, (short)0, acc[n], false, false);
        }
    }

    // C/D layout: VGPR v -> row = 8*half + v, col = l16
    const int rbase = blockIdx.x * 128 + wave * 16 + half * 8;
    if (blockIdx.x * 128 + 128 <= nrows) {
        // full tile: unguarded coalesced stores
#pragma unroll
        for (int n = 0; n < 8; ++n)
#pragma unroll
            for (int v = 0; v < 8; ++v)
                OUT[(size_t)(rbase + v) * FEAT + n * 16 + l16] = acc[n][v];
    } else {
#pragma unroll
        for (int n = 0; n < 8; ++n)
#pragma unroll
            for (int v = 0; v < 8; ++v)
                if (rbase + v < nrows)
                    OUT[(size_t)(rbase + v) * FEAT + n * 16 + l16] = acc[n][v];
    }
}

// ---------------- pooling ----------------
__global__ void __launch_bounds__(FEAT)
k_pool(const float* __restrict__ h, const int* __restrict__ batch,
       float* __restrict__ addp, unsigned int* __restrict__ maxp,
       float* __restrict__ cnt, int n) {
    int f = threadIdx.x;
    for (int node = blockIdx.x; node < n; node += gridDim.x) {
        int g = batch[node];
        float v = h[(size_t)node * FEAT + f];
        atomicAdd(&addp[g * FEAT + f], v);
        // v >= 0 after ReLU -> u32 compare == float compare
        atomicMax(&maxp[g * FEAT + f], __float_as_uint(v));
        if (f == 0) atomicAdd(&cnt[g], 1.0f);
    }
}

__global__ void __launch_bounds__(FEAT)
k_enc(const float* __restrict__ addp, const unsigned int* __restrict__ maxp,
      const float* __restrict__ cnt, float* __restrict__ enc) {
    int g = blockIdx.x;
    int f = threadIdx.x;
    float s = addp[g * FEAT + f];
    float c = fmaxf(cnt[g], 1.0f);
    float m = __uint_as_float(maxp[g * FEAT + f]);
    enc[g * 384 + f]       = s;
    enc[g * 384 + 128 + f] = s / c;
    enc[g * 384 + 256 + f] = m;
}

// ---------------- MLP head (tiny: 64x384 @ 384x128, 64x128 @ 128x2) ----------------
__global__ void k_lin1(const float* __restrict__ enc, const float* __restrict__ w,
                       const float* __restrict__ b, float* __restrict__ h1) {
    int t = blockIdx.x * blockDim.x + threadIdx.x;  // 64*128 outputs
    if (t < 64 * FEAT) {
        int g = t >> 7, j = t & (FEAT - 1);
        float acc = b[j];
        for (int k = 0; k < 384; ++k)
            acc = fmaf(enc[g * 384 + k], w[k * FEAT + j], acc);
        h1[t] = fmaxf(acc, 0.0f);
    }
}

__global__ void k_lin2(const float* __restrict__ h1, const float* __restrict__ w,
                       const float* __restrict__ b, float* __restrict__ out) {
    int t = blockIdx.x * blockDim.x + threadIdx.x;  // 64*2 outputs
    if (t < 64 * 2) {
        int g = t >> 1, j = t & 1;
        float acc = b[j];
        for (int k = 0; k < FEAT; ++k)
            acc = fmaf(h1[g * FEAT + k], w[k * 2 + j], acc);
        out[t] = acc;
    }
}

// ---------------- host launch ----------------
extern "C" void kernel_launch(void* const* d_in, const int* in_sizes, int n_in,
                              void* d_out, int out_size, void* d_ws, size_t ws_size,
                              hipStream_t stream) {
    const float* x      = (const float*)d_in[0];
    const int*   ei     = (const int*)d_in[1];   // [2, E] src row then dst row
    const int*   batch  = (const int*)d_in[2];
    const float* conv_w = (const float*)d_in[3]; // [3,128,128]
    const float* conv_b = (const float*)d_in[4]; // [3,128]
    const float* lin1_w = (const float*)d_in[5];
    const float* lin1_b = (const float*)d_in[6];
    const float* lin2_w = (const float*)d_in[7];
    const float* lin2_b = (const float*)d_in[8];
    float* out = (float*)d_out;

    const int N = in_sizes[0] / FEAT;  // 50000
    const int E = in_sizes[1] / 2;     // 800000
    const int G = 64;

    const int* src = ei;
    const int* dst = ei + E;

    // workspace carve-out (all f32)
    float* ws   = (float*)d_ws;
    float* deg  = ws;  ws += N;
    float* dinv = ws;  ws += N;
    float* nrm  = ws;  ws += E;
    float* hw   = ws;  ws += (size_t)N * FEAT;   // GEMM result per layer
    float* hbuf = ws;  ws += (size_t)N * FEAT;   // aggregated/ReLU'd features
    float* addp = ws;  ws += G * FEAT;
    float* maxp = ws;  ws += G * FEAT;           // used as u32 bits
    float* cnt  = ws;  ws += G;
    float* h1   = ws;  ws += G * FEAT;

    float* enc = out + G * 2;  // d_out = [out 64x2 | enc 64x384]

    const int T = 256;
    const int nTotal = N * FEAT;

    // degree (self-loop included via init=1) -> dinv -> edge norms
    k_fill<<<(N + T - 1) / T, T, 0, stream>>>(deg, 1.0f, N);
    k_deg_edges<<<(E + T - 1) / T, T, 0, stream>>>(dst, deg, E);
    k_dinv<<<(N + T - 1) / T, T, 0, stream>>>(deg, dinv, N);
    k_norm<<<(E + T - 1) / T, T, 0, stream>>>(src, dst, dinv, nrm, E);

    const int gemmBlocks = (N + 127) / 128;  // 391
    for (int l = 0; l < 3; ++l) {
        const float* hin = (l == 0) ? x : hbuf;
        k_gemm128<<<gemmBlocks, 256, 0, stream>>>(hin, conv_w + (size_t)l * FEAT * FEAT, hw, N);
        k_bias_selfloop<<<(nTotal + T - 1) / T, T, 0, stream>>>(hw, dinv, conv_b + l * FEAT, hbuf, nTotal);
        k_scatter<<<4096, FEAT, 0, stream>>>(hw, src, dst, nrm, hbuf, E);
        k_relu<<<(nTotal + T - 1) / T, T, 0, stream>>>(hbuf, nTotal);
    }

    // pooling (addp/maxp/cnt zero-init; maxp bits 0 == +0.0f, valid since h>=0)
    k_fill<<<(G * FEAT + T - 1) / T, T, 0, stream>>>(addp, 0.0f, G * FEAT);
    k_fill<<<(G * FEAT + T - 1) / T, T, 0, stream>>>(maxp, 0.0f, G * FEAT);
    k_fill<<<1, G, 0, stream>>>(cnt, 0.0f, G);
    k_pool<<<2048, FEAT, 0, stream>>>(hbuf, batch, addp, (unsigned int*)maxp, cnt, N);
    k_enc<<<G, FEAT, 0, stream>>>(addp, (const unsigned int*)maxp, cnt, enc);

    // MLP head
    k_lin1<<<(G * FEAT + T - 1) / T, T, 0, stream>>>(enc, lin1_w, lin1_b, h1);
    k_lin2<<<1, G * 2, 0, stream>>>(h1, lin2_w, lin2_b, out);
}